// DASAndPixelInterpolator_65901978189953
// MI455X (gfx1250) — compile-verified
//
#include <hip/hip_runtime.h>

typedef float v2f __attribute__((ext_vector_type(2)));
typedef float v8f __attribute__((ext_vector_type(8)));

#define NXg 256
#define NYg 256
#define NEg 128
#define NTg 4096
#define NBg 8
#define DAS_ELEMS (NBg * NYg * NXg) /* 524288 */

// ---------------------------------------------------------------------------
// Kernel 0: init per-batch max accumulator (ws is poisoned to 0xAA by harness)
// ---------------------------------------------------------------------------
__global__ void init_max_kernel(int* __restrict__ maxbuf) {
    if (threadIdx.x < NBg) maxbuf[threadIdx.x] = 0;
}

// ---------------------------------------------------------------------------
// Kernel 1: gather + weight + streamed pixel_interp store + WMMA reduction.
// One wave (32 lanes) handles a 16-pixel x-tile of one row of one batch.
//   A matrix (16x4 f32):  lanes 0-15 hold K=0,1 ; lanes 16-31 hold K=2,3
//   B matrix = all ones  -> D[m,n] = sum_k A[m,k]  (element-sum per pixel)
// ---------------------------------------------------------------------------
__global__ __launch_bounds__(256) void das_wmma_kernel(
    const float* __restrict__ sino,   // (B, 1, E, T) f32
    const int*   __restrict__ tidx,   // (NY, NX, E) i32
    const float* __restrict__ wts,    // (NY, NX, E) f32
    float*       __restrict__ out,    // das (B,1,NY,NX) then pixel_interp (B,E,NY,NX)
    int*         __restrict__ maxbuf) // per-batch max (float bits)
{
    const int lane = threadIdx.x & 31;
    const int wave = threadIdx.x >> 5;
    const int wgid = blockIdx.x * 8 + wave;

    const int tilesPerImage = (NXg / 16) * NYg;   // 4096
    const int b   = wgid / tilesPerImage;
    const int rem = wgid - b * tilesPerImage;
    const int y   = rem >> 4;
    const int x0  = (rem & 15) << 4;

    const int p     = lane & 15;   // pixel within tile (A-matrix row M)
    const int khalf = lane >> 4;   // 0: K=0,1   1: K=2,3
    const int x     = x0 + p;

    const int pixBase = ((y << 8) + x) << 7;      // (y*NX + x) * NE
    const float* __restrict__ sino_b = sino + b * (NEg * NTg);
    float* __restrict__ pi_out = out + DAS_ELEMS;

    v8f c = {};                                    // f32 accumulator tile
    v2f bone; bone.x = 1.0f; bone.y = 1.0f;        // all-ones B matrix

    #pragma unroll 4
    for (int k = 0; k < NEg; k += 4) {
        const int e0 = k + (khalf << 1);
        const int e1 = e0 + 1;

        int i0 = tidx[pixBase + e0];
        int i1 = tidx[pixBase + e1];
        i0 = (i0 < 0) ? 0 : ((i0 > NTg - 1) ? NTg - 1 : i0);
        i1 = (i1 < 0) ? 0 : ((i1 > NTg - 1) ? NTg - 1 : i1);

        const float w0 = wts[pixBase + e0];
        const float w1 = wts[pixBase + e1];

        const float wv0 = sino_b[(e0 << 12) + i0] * w0;   // gather hits L2 (16MB sinogram)
        const float wv1 = sino_b[(e1 << 12) + i1] * w1;

        // Streaming 268MB output: non-temporal so L2 keeps geometry/sinogram.
        __builtin_nontemporal_store(wv0, &pi_out[(((b << 7) + e0) << 16) + (y << 8) + x]);
        __builtin_nontemporal_store(wv1, &pi_out[(((b << 7) + e1) << 16) + (y << 8) + x]);

        v2f a; a.x = wv0; a.y = wv1;
        c = __builtin_amdgcn_wmma_f32_16x16x4_f32(
                /*neg_a=*/false, a, /*neg_b=*/false, bone,
                /*c_mod=*/(short)0, c, /*reuse_a=*/false, /*reuse_b=*/false);
    }

    // Extract pixel sums from D layout: VGPR r -> M=r (lanes 0-15) / M=r+8 (lanes 16-31).
    // All 16 columns identical (B = ones), so any column works.
    const int r = lane & 7;
    float v = c[0];
    v = (r == 1) ? c[1] : v;
    v = (r == 2) ? c[2] : v;
    v = (r == 3) ? c[3] : v;
    v = (r == 4) ? c[4] : v;
    v = (r == 5) ? c[5] : v;
    v = (r == 6) ? c[6] : v;
    v = (r == 7) ? c[7] : v;

    const bool storing = (lane < 8) || (lane >= 16 && lane < 24);
    const int  pix     = (lane < 16) ? r : (8 + r);
    const float dval   = fmaxf(v, 0.0f);

    // Wave-level max reduce (values >= 0), then one atomic per wave.
    float red = storing ? dval : 0.0f;
    red = fmaxf(red, __shfl_xor(red, 16, 32));
    red = fmaxf(red, __shfl_xor(red, 8, 32));
    red = fmaxf(red, __shfl_xor(red, 4, 32));
    red = fmaxf(red, __shfl_xor(red, 2, 32));
    red = fmaxf(red, __shfl_xor(red, 1, 32));
    if (lane == 0) atomicMax(&maxbuf[b], __float_as_int(red)); // exact for floats >= 0

    if (storing) out[(b * NYg + y) * NXg + x0 + pix] = dval;   // unnormalized das
}

// ---------------------------------------------------------------------------
// Kernel 2: normalize das in place by per-batch max (clamp-norm semantics).
// ---------------------------------------------------------------------------
__global__ __launch_bounds__(256) void das_norm_kernel(
    float* __restrict__ out_das, const int* __restrict__ maxbuf)
{
    const int i = blockIdx.x * 256 + threadIdx.x;   // 0 .. DAS_ELEMS-1
    const int b = i >> 16;                          // NY*NX = 65536 per batch
    float m = __int_as_float(maxbuf[b]);
    m = (m > 1e-8f) ? m : 1.0f;
    out_das[i] = out_das[i] / m;
}

extern "C" void kernel_launch(void* const* d_in, const int* in_sizes, int n_in,
                              void* d_out, int out_size, void* d_ws, size_t ws_size,
                              hipStream_t stream) {
    const float* sino = (const float*)d_in[0];   // (8,1,128,4096) f32
    const int*   tidx = (const int*)  d_in[1];   // (256,256,128) i32
    const float* wts  = (const float*)d_in[2];   // (256,256,128) f32
    float* out = (float*)d_out;                  // 524288 das + 67108864 pixel_interp
    int* maxbuf = (int*)d_ws;                    // 8 ints

    init_max_kernel<<<1, 32, 0, stream>>>(maxbuf);

    // 8 batches * 256 rows * 16 x-tiles = 32768 waves / 8 waves per block
    das_wmma_kernel<<<4096, 256, 0, stream>>>(sino, tidx, wts, out, maxbuf);

    das_norm_kernel<<<DAS_ELEMS / 256, 256, 0, stream>>>(out, maxbuf);
}